// Model_21423296872760
// MI455X (gfx1250) — compile-verified
//
#include <hip/hip_runtime.h>
#include <hip/hip_bf16.h>
#include <stdint.h>

#define N_OSC 2048
#define N_SAMPLES 16384
#define LAT 32

typedef __attribute__((ext_vector_type(2))) float v2f;
typedef __attribute__((ext_vector_type(8))) float v8f;
typedef uint32_t u32x4 __attribute__((ext_vector_type(4)));
typedef uint32_t u32x8 __attribute__((ext_vector_type(8)));

// v_sin_f32 computes sin(2*pi*x) (ISA-normalized input). SIN01(x) = sin(2*pi*x).
#if __has_builtin(__builtin_amdgcn_sinf)
  #define SIN01(x) __builtin_amdgcn_sinf(x)
#else
  #define SIN01(x) __sinf(6.28318530717958647f * (x))
#endif

// ---------------------------------------------------------------------------
// Kernel 1: abs-copy latent into the freq_params / amp_params output slots.
// d_out layout: [0,16384) signal | [16384,81920) freq | [81920,147456) amp
// ---------------------------------------------------------------------------
__global__ __launch_bounds__(256) void absparams_kernel(const float* __restrict__ latent,
                                                        float* __restrict__ out) {
    int i = blockIdx.x * 256 + threadIdx.x;           // 0 .. 65535
    out[N_SAMPLES + i]               = fabsf(latent[N_OSC * LAT + i]); // freq_params
    out[N_SAMPLES + N_OSC * LAT + i] = fabsf(latent[i]);               // amp_params
}

// ---------------------------------------------------------------------------
// Kernel 2: per-oscillator prefix sums of |freq latent| -> Spre[osc*32+m]
// ---------------------------------------------------------------------------
__global__ __launch_bounds__(256) void prefix_kernel(const float* __restrict__ latent,
                                                     float* __restrict__ Spre) {
    int osc = blockIdx.x * 256 + threadIdx.x;         // 0 .. 2047
    const float* f = latent + N_OSC * LAT + osc * LAT;
    float s = 0.0f;
    #pragma unroll
    for (int m = 0; m < LAT; ++m) {
        s += fabsf(f[m]);
        Spre[osc * LAT + m] = s;
    }
}

// ---------------------------------------------------------------------------
// Kernel 3: main synthesis. One block (8 waves) per 16-sample tile (block-
// uniform segment k). The Tensor Data Mover gathers the 5 needed per-
// oscillator columns (strided tile 1x2048 of a 32x2048 tensor) into LDS;
// a branchless repack bakes {Pc,fk,dfk,ak,dak}; the hot loop is pure
// ds_load -> fma/v_sin -> V_WMMA_F32_16X16X4_F32 (A = ones column reducer).
// ---------------------------------------------------------------------------
__global__ __launch_bounds__(256) void synth_kernel(const float* __restrict__ latent,
                                                    const float* __restrict__ Spre,
                                                    float* __restrict__ out) {
    __shared__ float  stage[5][N_OSC];  // S, f_lo, f_hi, a_lo, a_hi   40 KB
    __shared__ float4 p4[N_OSC];        // {Pc, fk, dfk, ak}           32 KB
    __shared__ float  pd[N_OSC];        // dak                          8 KB
    __shared__ float  red[8 * 16];

    const int tid  = threadIdx.x;
    const int lane = tid & 31;
    const int wave = tid >> 5;
    const int t0   = blockIdx.x * 16;                 // tile base sample
    const int t    = t0 + (lane & 15);                // this lane's sample (N col)

    // Block-uniform region / segment (tiles never straddle 512-boundaries).
    const int region = (t0 < 256) ? 0 : ((t0 >= N_SAMPLES - 256) ? 2 : 1);
    const int kk     = (region == 0) ? 0 : ((region == 2) ? 31 : ((t0 - 256) >> 9));
    // Column choices: flat regions read the same column twice -> dfk,dak == 0.
    const int c0 = (region == 1) ? kk : ((region == 0) ? 0 : 31);
    const int c1 = (region == 1) ? (kk + 1) : c0;
    const int cS = (kk > 0) ? (kk - 1) : 0;
    const float Smul = (region != 0 && kk > 0) ? 512.0f : 0.0f;
    const float Pmul = (region == 0) ? 0.0f : 256.0f;

    // ---- TDM: one wave issues 5 column gathers (TENSORcnt-tracked) --------
    if (wave == 0) {
        const uint64_t baseF = (uint64_t)(uintptr_t)(latent + N_OSC * LAT);
        const uint64_t baseA = (uint64_t)(uintptr_t)latent;
        const uint64_t baseS = (uint64_t)(uintptr_t)Spre;
        const uint64_t ga[5] = { baseS + 4ull * (uint32_t)cS,
                                 baseF + 4ull * (uint32_t)c0,
                                 baseF + 4ull * (uint32_t)c1,
                                 baseA + 4ull * (uint32_t)c0,
                                 baseA + 4ull * (uint32_t)c1 };
        #pragma unroll
        for (int c = 0; c < 5; ++c) {
            const uint32_t lds_off = (uint32_t)(uintptr_t)&stage[c][0];
            u32x4 g0;
            g0[0] = 1u;                                  // count=1, user mode
            g0[1] = lds_off;                             // lds_addr
            g0[2] = (uint32_t)ga[c];                     // global_addr[31:0]
            g0[3] = (uint32_t)((ga[c] >> 32) & 0x01FFFFFFull) | 0x80000000u; // type=2
            u32x8 g1;
            g1[0] = 0x00020000u;     // workgroup_mask=0, data_size=4B, no flags
            g1[1] = (uint32_t)LAT << 16;     // tensor_dim0 = 32 (lo16)
            g1[2] = (uint32_t)N_OSC << 16;   // tensor_dim0 hi=0 | tensor_dim1 lo16
            g1[3] = 1u << 16;                // tensor_dim1 hi=0 | tile_dim0 = 1
            g1[4] = (uint32_t)N_OSC;         // tile_dim1 = 2048 | tile_dim2 = 0
            g1[5] = (uint32_t)LAT;           // tensor_dim0_stride = 32
            g1[6] = 0u;                      // stride hi | dim1_stride lo (unused, 2D)
            g1[7] = 0u;
            asm volatile("tensor_load_to_lds %0, %1" :: "s"(g0), "s"(g1) : "memory");
        }
        __builtin_amdgcn_s_wait_tensorcnt(0);
    }
    __syncthreads();

    // ---- branchless repack: 8 oscillators / thread -------------------------
    #pragma unroll
    for (int i = 0; i < 8; ++i) {
        const int osc = i * 256 + tid;
        const float S  = stage[0][osc];          // prefix sum (already |.|-summed)
        const float fk = fabsf(stage[1][osc]);
        const float f1 = fabsf(stage[2][osc]);
        const float a0 = fabsf(stage[3][osc]);
        const float a1 = fabsf(stage[4][osc]);
        p4[osc] = make_float4(Smul * S + Pmul * fk, fk,
                              (f1 - fk) * (1.0f / 1024.0f), a0);
        pd[osc] = a1 - a0;
    }
    __syncthreads();

    // ---- per-lane sample constants ----
    float q;
    if (region == 0)      q = (float)(t + 1);
    else if (region == 2) q = (float)(t - (N_SAMPLES - 256) + 1);
    else                  q = (float)(t - 256 - (kk << 9) + 1);
    const float q2 = q * q;
    const float wq = (region == 1)
                   ? ((float)(t - 256 - (kk << 9)) + 0.5f) * (1.0f / 512.0f)
                   : 0.0f;

    v8f c = {};
    v2f ones; ones.x = 1.0f; ones.y = 1.0f;
    const int obase = wave * 256;

    // ---- hot loop: branch-free, EXEC all-ones throughout -------------------
    #pragma unroll 2
    for (int s = 0; s < 64; ++s) {
        // B is 4x16 (K x N): lanes 0-15 hold K=0 (b.x) / K=1 (b.y),
        // lanes 16-31 hold K=2 (b.x) / K=3 (b.y). K indexes 4 oscillators.
        const int o = obase + s * 4 + ((lane >> 4) << 1);
        v2f b;
        #pragma unroll
        for (int u = 0; u < 2; ++u) {
            const float4 v  = p4[o + u];          // ds_load_b128
            const float dak = pd[o + u];          // ds_load_b32
            const float cum = v.x + q * v.y + q2 * v.z;   // phase/pi
            const float amp = v.w + dak * wq;
            const float h   = 0.5f * cum;
            const float r   = h - floorf(h);      // frac(cum/2) in [0,1)
            const float sv  = SIN01(r) * amp;     // sin(pi*cum)*amp
            if (u == 0) b.x = sv; else b.y = sv;
        }
        // D = ones(16x4) * B(4x16) + C : every row of D = 16 column sums.
        c = __builtin_amdgcn_wmma_f32_16x16x4_f32(
                false, ones, false, b, (short)0, c, false, false);
    }

    // D row M=0 lives in c[0] on lanes 0-15 (N = lane).
    if (lane < 16) red[wave * 16 + lane] = c[0];
    __syncthreads();

    if (tid < 16) {
        float s = 0.0f;
        #pragma unroll
        for (int w = 0; w < 8; ++w) s += red[w * 16 + tid];  // fixed order
        out[t0 + tid] = s * (1.0f / (float)N_OSC);
    }
}

// ---------------------------------------------------------------------------
extern "C" void kernel_launch(void* const* d_in, const int* in_sizes, int n_in,
                              void* d_out, int out_size, void* d_ws, size_t ws_size,
                              hipStream_t stream) {
    (void)in_sizes; (void)n_in; (void)out_size; (void)ws_size;
    const float* latent = (const float*)d_in[1];   // d_in[0] = x (unused)
    float* out  = (float*)d_out;
    float* Spre = (float*)d_ws;                    // needs 2048*32*4 = 256 KB

    prefix_kernel<<<N_OSC / 256, 256, 0, stream>>>(latent, Spre);
    absparams_kernel<<<(N_OSC * LAT) / 256, 256, 0, stream>>>(latent, out);
    synth_kernel<<<N_SAMPLES / 16, 256, 0, stream>>>(latent, Spre, out);
}